// GraphSAGEClassifier_76536317214877
// MI455X (gfx1250) — compile-verified
//
#include <hip/hip_runtime.h>
#include <hip/hip_bf16.h>

typedef __attribute__((ext_vector_type(2))) float v2f;
typedef __attribute__((ext_vector_type(8))) float v8f;

#define TPB 256

// ---------------------------------------------------------------- zero fill
__global__ __launch_bounds__(TPB) void zero_f32(float* __restrict__ p, long long n) {
    long long i = (long long)blockIdx.x * TPB + threadIdx.x;
    long long stride = (long long)gridDim.x * TPB;
    for (; i < n; i += stride) p[i] = 0.0f;
}

// ---------------------------------------------------------------- degree (once)
__global__ __launch_bounds__(TPB) void degree_kernel(const int* __restrict__ dst,
                                                     float* __restrict__ deg,
                                                     int nEdges, int nNodes) {
    int e = blockIdx.x * TPB + threadIdx.x;
    if (e >= nEdges) return;
    int d = dst[e];
    if ((unsigned)d < (unsigned)nNodes) atomicAdd(&deg[d], 1.0f);
}

// ---------------------------------------------------------------- edge scatter
// thread idx = e*16 + cg ; 16 lanes cover one edge's 64 channels (float4 each)
__global__ __launch_bounds__(TPB) void scatter_add_kernel(const float* __restrict__ x,
                                                          const int* __restrict__ src,
                                                          const int* __restrict__ dst,
                                                          float* __restrict__ agg,
                                                          int nEdges, int nNodes) {
    long long idx = (long long)blockIdx.x * TPB + threadIdx.x;
    long long total = (long long)nEdges * 16;
    if (idx >= total) return;
    int e  = (int)(idx >> 4);
    int cg = ((int)idx & 15) << 2;
    int s = src[e];
    int d = dst[e];
    if ((unsigned)s >= (unsigned)nNodes || (unsigned)d >= (unsigned)nNodes) return;
    const float4 v = *(const float4*)(x + (long long)s * 64 + cg);
    float* ap = agg + (long long)d * 64 + cg;
    atomicAdd(ap + 0, v.x);
    atomicAdd(ap + 1, v.y);
    atomicAdd(ap + 2, v.z);
    atomicAdd(ap + 3, v.w);
}

// ---------------------------------------------------------------- fused SAGE update
// out = relu( [agg/deg | x] @ [Wl ; Wr] + b )   via V_WMMA_F32_16X16X4_F32
// block = 256 threads = 8 waves; block covers 32 rows x 64 cols
// wave (rtile 0..1, ctile 0..3) computes one 16x16 output tile, K = 128.
#define A_STRIDE 132   // padded LDS row stride to break bank conflicts

__global__ __launch_bounds__(TPB) void sage_update_kernel(const float* __restrict__ agg,
                                                          const float* __restrict__ deg,
                                                          const float* __restrict__ xin,
                                                          const float* __restrict__ Wl,
                                                          const float* __restrict__ Wr,
                                                          const float* __restrict__ bias,
                                                          float* __restrict__ out,
                                                          int nNodes) {
    __shared__ float Bs[128 * 64];        // concat weights [Wl ; Wr], 32 KB
    __shared__ float As[32 * A_STRIDE];   // 32-row A tile (scaled agg | x)

    const int tid = threadIdx.x;
    const int rowBase = blockIdx.x * 32;

    // stage weights: Bs[k][n]
    for (int i = tid; i < 128 * 64; i += TPB) {
        int k = i >> 6, n = i & 63;
        Bs[i] = (k < 64) ? Wl[k * 64 + n] : Wr[(k - 64) * 64 + n];
    }
    // stage A tile: As[r][k] = k<64 ? agg[row][k]/max(deg,1) : x[row][k-64]
    for (int i = tid; i < 32 * 128; i += TPB) {
        int r = i >> 7, k = i & 127;
        int row = rowBase + r;
        float v = 0.0f;
        if (row < nNodes) {
            if (k < 64) {
                float inv = 1.0f / fmaxf(deg[row], 1.0f);
                v = agg[(long long)row * 64 + k] * inv;
            } else {
                v = xin[(long long)row * 64 + (k - 64)];
            }
        }
        As[r * A_STRIDE + k] = v;
    }
    __syncthreads();

    const int wave  = tid >> 5;
    const int lane  = tid & 31;
    const int half  = lane >> 4;   // half-wave: K offset +2 (A/B), M offset +8 (C/D)
    const int lrow  = lane & 15;
    const int rtile = wave >> 2;   // 0..1
    const int ctile = wave & 3;    // 0..3
    const int colBase = ctile * 16;
    const int aRow    = rtile * 16 + lrow;

    // C/D layout: VGPR i, lanes 0-15 -> M=i, lanes 16-31 -> M=i+8; N = lrow.
    // bias depends on N only -> preload into accumulator.
    v8f acc;
    float bv = bias[colBase + lrow];
#pragma unroll
    for (int i = 0; i < 8; ++i) acc[i] = bv;

#pragma unroll
    for (int kk = 0; kk < 32; ++kk) {
        int k0 = kk * 4 + half * 2;
        v2f a, b;
        // A 16x4 frag: VGPR0 = K=k0 (K=k0+2 upper half), VGPR1 = K=k0+1 (k0+3)
        a.x = As[aRow * A_STRIDE + k0];
        a.y = As[aRow * A_STRIDE + k0 + 1];
        // B 4x16 frag: row K striped across lanes within a VGPR
        b.x = Bs[k0 * 64 + colBase + lrow];
        b.y = Bs[(k0 + 1) * 64 + colBase + lrow];
        acc = __builtin_amdgcn_wmma_f32_16x16x4_f32(
            /*neg_a=*/false, a, /*neg_b=*/false, b,
            /*c_mod=*/(short)0, acc, /*reuse_a=*/false, /*reuse_b=*/false);
    }

    const int outRowBase = rowBase + rtile * 16;
    const int col = colBase + lrow;
#pragma unroll
    for (int r = 0; r < 8; ++r) {
        int row = outRowBase + r + half * 8;
        if (row < nNodes)
            out[(long long)row * 64 + col] = fmaxf(acc[r], 0.0f);
    }
}

// ---------------------------------------------------------------- graph pooling
__global__ __launch_bounds__(TPB) void pool_kernel(const float* __restrict__ h,
                                                   const int* __restrict__ batch,
                                                   float* __restrict__ sums,
                                                   float* __restrict__ cnts,
                                                   int nNodes, int nGraphs) {
    long long idx = (long long)blockIdx.x * TPB + threadIdx.x;
    long long total = (long long)nNodes * 16;
    if (idx >= total) return;
    int nd = (int)(idx >> 4);
    int cg = ((int)idx & 15) << 2;
    int b = batch[nd];
    if ((unsigned)b >= (unsigned)nGraphs) return;
    const float4 v = *(const float4*)(h + (long long)nd * 64 + cg);
    float* sp = sums + (long long)b * 64 + cg;
    atomicAdd(sp + 0, v.x);
    atomicAdd(sp + 1, v.y);
    atomicAdd(sp + 2, v.z);
    atomicAdd(sp + 3, v.w);
    if (cg == 0) atomicAdd(&cnts[b], 1.0f);
}

// ---------------------------------------------------------------- MLP head (tiny)
__global__ __launch_bounds__(TPB) void head_kernel(const float* __restrict__ sums,
                                                   const float* __restrict__ cnts,
                                                   const float* __restrict__ Wc1,
                                                   const float* __restrict__ bc1,
                                                   const float* __restrict__ Wc2,
                                                   const float* __restrict__ bc2,
                                                   float* __restrict__ out) {
    __shared__ float g[64 * 64];
    __shared__ float hid[64 * 32];
    int tid = threadIdx.x;
    for (int i = tid; i < 64 * 64; i += TPB) {
        int gi = i >> 6;
        g[i] = sums[i] / fmaxf(cnts[gi], 1.0f);
    }
    __syncthreads();
    for (int i = tid; i < 64 * 32; i += TPB) {
        int gi = i >> 5, j = i & 31;
        float acc = bc1[j];
#pragma unroll
        for (int k = 0; k < 64; ++k) acc += g[gi * 64 + k] * Wc1[k * 32 + j];
        hid[i] = fmaxf(acc, 0.0f);
    }
    __syncthreads();
    for (int i = tid; i < 64 * 2; i += TPB) {
        int gi = i >> 1, j = i & 1;
        float acc = bc2[j];
#pragma unroll
        for (int k = 0; k < 32; ++k) acc += hid[gi * 32 + k] * Wc2[k * 2 + j];
        out[i] = acc;
    }
}

// ---------------------------------------------------------------- launch
extern "C" void kernel_launch(void* const* d_in, const int* in_sizes, int n_in,
                              void* d_out, int out_size, void* d_ws, size_t ws_size,
                              hipStream_t stream) {
    const float* x     = (const float*)d_in[0];
    const int*   eidx  = (const int*)d_in[1];
    const int*   batch = (const int*)d_in[2];
    const float* W1l = (const float*)d_in[3];
    const float* b1  = (const float*)d_in[4];
    const float* W1r = (const float*)d_in[5];
    const float* W2l = (const float*)d_in[6];
    const float* b2  = (const float*)d_in[7];
    const float* W2r = (const float*)d_in[8];
    const float* W3l = (const float*)d_in[9];
    const float* b3  = (const float*)d_in[10];
    const float* W3r = (const float*)d_in[11];
    const float* Wc1 = (const float*)d_in[12];
    const float* bc1 = (const float*)d_in[13];
    const float* Wc2 = (const float*)d_in[14];
    const float* bc2 = (const float*)d_in[15];
    float* out = (float*)d_out;

    const int nNodes  = in_sizes[0] / 64;
    const int nEdges  = in_sizes[1] / 2;
    const int nGraphs = 64;
    const int* src = eidx;
    const int* dst = eidx + nEdges;

    // workspace layout
    const long long featN = (long long)nNodes * 64;
    float* agg  = (float*)d_ws;
    float* h0   = agg + featN;
    float* h1   = h0 + featN;
    float* deg  = h1 + featN;
    float* sums = deg + nNodes;
    float* cnts = sums + (long long)nGraphs * 64;

    const int zgrid = 4096;
    const int egrid = (int)(((long long)nEdges * 16 + TPB - 1) / TPB);
    const int dgrid = (nEdges + TPB - 1) / TPB;
    const int ugrid = (nNodes + 31) / 32;
    const int pgrid = (int)(((long long)nNodes * 16 + TPB - 1) / TPB);

    // degree (graph is fixed across layers; recomputed each call for determinism)
    zero_f32<<<zgrid, TPB, 0, stream>>>(deg, nNodes);
    degree_kernel<<<dgrid, TPB, 0, stream>>>(dst, deg, nEdges, nNodes);

    // layer 1: x -> h0
    zero_f32<<<zgrid, TPB, 0, stream>>>(agg, featN);
    scatter_add_kernel<<<egrid, TPB, 0, stream>>>(x, src, dst, agg, nEdges, nNodes);
    sage_update_kernel<<<ugrid, TPB, 0, stream>>>(agg, deg, x, W1l, W1r, b1, h0, nNodes);

    // layer 2: h0 -> h1
    zero_f32<<<zgrid, TPB, 0, stream>>>(agg, featN);
    scatter_add_kernel<<<egrid, TPB, 0, stream>>>(h0, src, dst, agg, nEdges, nNodes);
    sage_update_kernel<<<ugrid, TPB, 0, stream>>>(agg, deg, h0, W2l, W2r, b2, h1, nNodes);

    // layer 3: h1 -> h0
    zero_f32<<<zgrid, TPB, 0, stream>>>(agg, featN);
    scatter_add_kernel<<<egrid, TPB, 0, stream>>>(h1, src, dst, agg, nEdges, nNodes);
    sage_update_kernel<<<ugrid, TPB, 0, stream>>>(agg, deg, h1, W3l, W3r, b3, h0, nNodes);

    // global mean pool + head
    zero_f32<<<1, TPB, 0, stream>>>(sums, (long long)nGraphs * 64 + nGraphs);
    pool_kernel<<<pgrid, TPB, 0, stream>>>(h0, batch, sums, cnts, nNodes, nGraphs);
    head_kernel<<<1, TPB, 0, stream>>>(sums, cnts, Wc1, bc1, Wc2, bc2, out);
}